// GAT_33663953666793
// MI455X (gfx1250) — compile-verified
//
#include <hip/hip_runtime.h>
#include <hip/hip_bf16.h>
#include <math.h>

typedef float v2f __attribute__((ext_vector_type(2)));
typedef float v8f __attribute__((ext_vector_type(8)));

// ---------------------------------------------------------------------------
// GEMM C[M,64] = A[M,K] @ B[K,64]   (fp32, V_WMMA_F32_16X16X4_F32)
// Block = 256 threads = 8 waves. B staged in LDS as k-pairs:
//   Bp[p][n] = {B[2p,n], B[2p+1,n]}  (float2, row stride 80 -> bank-conflict-free)
// Each wave computes a 16x64 strip: one A b64 load + 4 ds b64 loads feed 4 WMMAs.
// f32 A 16x4 layout: lanes 0-15 hold K={k,k+1}, lanes 16-31 K={k+2,k+3};
//   v0 = A[m, k+2*hi], v1 = A[m, k+1+2*hi]   (hi = lane>>4, m = tileM + (lane&15))
// B 4x16 mirrors A with N across lanes -> operand = Bp[k/2 + hi][n] directly.
// C/D: vgpr g -> row g + 8*hi, col = lane&15
// ---------------------------------------------------------------------------
#define BP_STRIDE 80
__global__ void gat_gemm_wmma_f32(const float* __restrict__ A,
                                  const float* __restrict__ B,
                                  float* __restrict__ C,
                                  int M, int K) {
    __shared__ v2f Bp[64 * BP_STRIDE];           // K<=128 -> 64 pairs, 40 KB
    const int tid = threadIdx.x;
    const int npairs = K >> 1;
    for (int i = tid; i < npairs * 64; i += 256) {
        int p = i >> 6, n = i & 63;
        v2f t;
        t.x = B[(size_t)(2 * p) * 64 + n];
        t.y = B[(size_t)(2 * p + 1) * 64 + n];
        Bp[p * BP_STRIDE + n] = t;
    }
    __syncthreads();

    const int wave  = blockIdx.x * 8 + (tid >> 5);
    const int tileM = wave << 4;
    if (tileM >= M) return;

    const int lane = tid & 31;
    const int l    = lane & 15;
    const int hi   = lane >> 4;

    const float* arow = A + (size_t)(tileM + l) * K;

    v8f acc0 = {}, acc1 = {}, acc2 = {}, acc3 = {};
    for (int k = 0; k < K; k += 4) {
        v2f a;
        a.x = arow[k + 2 * hi];                  // contiguous pair -> b64 load
        a.y = arow[k + 2 * hi + 1];
        const v2f* brow = &Bp[((k >> 1) + hi) * BP_STRIDE + l];
        v2f bA = brow[0];                        // ds_load_b64, even VGPR pair
        v2f bB = brow[16];
        v2f bC = brow[32];
        v2f bD = brow[48];
        acc0 = __builtin_amdgcn_wmma_f32_16x16x4_f32(false, a, false, bA, (short)0, acc0, false, false);
        acc1 = __builtin_amdgcn_wmma_f32_16x16x4_f32(false, a, false, bB, (short)0, acc1, false, false);
        acc2 = __builtin_amdgcn_wmma_f32_16x16x4_f32(false, a, false, bC, (short)0, acc2, false, false);
        acc3 = __builtin_amdgcn_wmma_f32_16x16x4_f32(false, a, false, bD, (short)0, acc3, false, false);
    }
#pragma unroll
    for (int g = 0; g < 8; ++g) {
        int row = tileM + g + 8 * hi;
        if (row < M) {
            float* crow = C + (size_t)row * 64;
            crow[l]      = acc0[g];
            crow[16 + l] = acc1[g];
            crow[32 + l] = acc2[g];
            crow[48 + l] = acc3[g];
        }
    }
}

// wdd[d] = sum_c W[d,c] * att[c]   (folds  a_d = x @ (W_dst @ att_dst))
__global__ void gat_matvec(const float* __restrict__ W, const float* __restrict__ att,
                           float* __restrict__ wdd, int D, int C) {
    int d = blockIdx.x * blockDim.x + threadIdx.x;
    if (d >= D) return;
    float s = 0.f;
    for (int c = 0; c < C; ++c) s += W[(size_t)d * C + c] * att[c];
    wdd[d] = s;
}

// agg = 0, emax = -inf, denom = 0
__global__ void gat_init_nodes(float* __restrict__ agg, float* __restrict__ emax,
                               float* __restrict__ denom, int N, int C) {
    int i = blockIdx.x * blockDim.x + threadIdx.x;
    if (i < N * C) agg[i] = 0.f;
    if (i < N) { emax[i] = -INFINITY; denom[i] = 0.f; }
}

// per-node logits: a_s[n] = xs[n,:]@att_src ; a_d[n] = xin[n,:]@wdd  (one wave/node)
__global__ void gat_node_logits(const float* __restrict__ xs, const float* __restrict__ atts,
                                int C, const float* __restrict__ xin,
                                const float* __restrict__ wdd, int D,
                                float* __restrict__ a_s, float* __restrict__ a_d, int N) {
    int node = blockIdx.x * (blockDim.x >> 5) + (threadIdx.x >> 5);
    int lane = threadIdx.x & 31;
    if (node >= N) return;
    float s = 0.f, d = 0.f;
    for (int c = lane; c < C; c += 32) s += xs[(size_t)node * C + c] * atts[c];
    for (int c = lane; c < D; c += 32) d += xin[(size_t)node * D + c] * wdd[c];
    for (int off = 16; off; off >>= 1) {
        s += __shfl_down(s, off);
        d += __shfl_down(d, off);
    }
    if (lane == 0) { a_s[node] = s; a_d[node] = d; }
}

__device__ inline void atomicMaxF32(float* addr, float val) {
    if (val >= 0.f) atomicMax((int*)addr, __float_as_int(val));
    else            atomicMin((unsigned int*)addr, __float_as_uint(val));
}

// e = leaky_relu(a_s[src]+a_d[dst]); segment-max into emax[dst]
__global__ void gat_edge_pass1(const int* __restrict__ src, const int* __restrict__ dst,
                               const float* __restrict__ a_s, const float* __restrict__ a_d,
                               float* __restrict__ ebuf, float* __restrict__ emax, int E) {
    int i = blockIdx.x * blockDim.x + threadIdx.x;
    if (i >= E) return;
    float e = a_s[src[i]] + a_d[dst[i]];
    e = (e > 0.f) ? e : 0.2f * e;
    ebuf[i] = e;
    atomicMaxF32(&emax[dst[i]], e);
}

// ex = exp(e - emax[dst]); denom[dst] += ex  (ebuf overwritten in place)
__global__ void gat_edge_pass2(const int* __restrict__ dst, float* __restrict__ ebuf,
                               const float* __restrict__ emax, float* __restrict__ denom, int E) {
    int i = blockIdx.x * blockDim.x + threadIdx.x;
    if (i >= E) return;
    int d = dst[i];
    float ex = __expf(ebuf[i] - emax[d]);
    ebuf[i] = ex;
    atomicAdd(&denom[d], ex);
}

// agg[dst] += alpha * xs[src] ; 16 lanes per edge, float4 per lane (C==64)
__global__ void gat_edge_pass3(const int* __restrict__ src, const int* __restrict__ dst,
                               const float* __restrict__ exbuf, const float* __restrict__ denom,
                               const float* __restrict__ xs, float* __restrict__ agg, int E) {
    int t = blockIdx.x * blockDim.x + threadIdx.x;
    int edge = t >> 4;
    int sub  = t & 15;
    if (edge >= E) return;
    int s = src[edge], d = dst[edge];
    float alpha = exbuf[edge] / (denom[d] + 1e-16f);
    float4 v = ((const float4*)(xs + (size_t)s * 64))[sub];
    float* o = agg + (size_t)d * 64 + sub * 4;
    atomicAdd(o + 0, alpha * v.x);
    atomicAdd(o + 1, alpha * v.y);
    atomicAdd(o + 2, alpha * v.z);
    atomicAdd(o + 3, alpha * v.w);
}

// out = (agg + bias) + (skip + bl)  [+ relu]
__global__ void gat_combine(const float* __restrict__ agg, const float* __restrict__ skip,
                            const float* __restrict__ bias, const float* __restrict__ bl,
                            float* __restrict__ out, int N, int C, int do_relu) {
    int i = blockIdx.x * blockDim.x + threadIdx.x;
    if (i >= N * C) return;
    int c = i & (C - 1);
    float v = agg[i] + skip[i] + bias[c] + bl[c];
    if (do_relu) v = fmaxf(v, 0.f);
    out[i] = v;
}

extern "C" void kernel_launch(void* const* d_in, const int* in_sizes, int n_in,
                              void* d_out, int out_size, void* d_ws, size_t ws_size,
                              hipStream_t stream) {
    const float* x        = (const float*)d_in[0];
    const int*   eidx     = (const int*)  d_in[1];
    const float* W_src1   = (const float*)d_in[3];
    const float* W_dst1   = (const float*)d_in[4];
    const float* att_src1 = (const float*)d_in[5];
    const float* att_dst1 = (const float*)d_in[6];
    const float* bias1    = (const float*)d_in[7];
    const float* Wl1      = (const float*)d_in[8];
    const float* bl1      = (const float*)d_in[9];
    const float* W_src2   = (const float*)d_in[10];
    const float* W_dst2   = (const float*)d_in[11];
    const float* att_src2 = (const float*)d_in[12];
    const float* att_dst2 = (const float*)d_in[13];
    const float* bias2    = (const float*)d_in[14];
    const float* Wl2      = (const float*)d_in[15];
    const float* bl2      = (const float*)d_in[16];

    const int DIN = 128, C = 64;
    const int N = in_sizes[0] / DIN;
    const int E = in_sizes[1] / 2;
    const int* src = eidx;
    const int* dst = eidx + E;

    // workspace carve (256B aligned), buffers reused across both layers
    char* w = (char*)d_ws;
    auto carve = [&](size_t bytes) -> float* {
        char* p = w; w += (bytes + 255) & ~(size_t)255; return (float*)p;
    };
    float* xs    = carve((size_t)N * C * 4);
    float* skip  = carve((size_t)N * C * 4);
    float* hbuf  = carve((size_t)N * C * 4);
    float* agg   = carve((size_t)N * C * 4);
    float* ebuf  = carve((size_t)E * 4);
    float* a_s   = carve((size_t)N * 4);
    float* a_d   = carve((size_t)N * 4);
    float* emax  = carve((size_t)N * 4);
    float* denom = carve((size_t)N * 4);
    float* wdd   = carve(512);

    auto run_layer = [&](const float* xin, int D,
                         const float* Wsrc, const float* Wdst,
                         const float* atts, const float* attd,
                         const float* bias, const float* Wl, const float* bl,
                         float* outbuf, int do_relu) {
        int NC = N * C;
        gat_init_nodes<<<(NC + 255) / 256, 256, 0, stream>>>(agg, emax, denom, N, C);
        gat_matvec<<<(D + 63) / 64, 64, 0, stream>>>(Wdst, attd, wdd, D, C);
        int mtiles  = (N + 15) / 16;                 // one wave per 16x64 strip
        int gblocks = (mtiles + 7) / 8;              // 8 waves per block
        gat_gemm_wmma_f32<<<gblocks, 256, 0, stream>>>(xin, Wsrc, xs,   N, D);
        gat_gemm_wmma_f32<<<gblocks, 256, 0, stream>>>(xin, Wl,   skip, N, D);
        gat_node_logits<<<(N + 7) / 8, 256, 0, stream>>>(xs, atts, C, xin, wdd, D, a_s, a_d, N);
        gat_edge_pass1<<<(E + 255) / 256, 256, 0, stream>>>(src, dst, a_s, a_d, ebuf, emax, E);
        gat_edge_pass2<<<(E + 255) / 256, 256, 0, stream>>>(dst, ebuf, emax, denom, E);
        long long t3 = (long long)E * 16;
        gat_edge_pass3<<<(int)((t3 + 255) / 256), 256, 0, stream>>>(src, dst, ebuf, denom, xs, agg, E);
        gat_combine<<<(NC + 255) / 256, 256, 0, stream>>>(agg, skip, bias, bl, outbuf, N, C, do_relu);
    };

    run_layer(x,    DIN, W_src1, W_dst1, att_src1, att_dst1, bias1, Wl1, bl1, hbuf, 1);
    run_layer(hbuf, C,   W_src2, W_dst2, att_src2, att_dst2, bias2, Wl2, bl2, (float*)d_out, 0);
}